// CRF_11751030522861
// MI455X (gfx1250) — compile-verified
//
#include <hip/hip_runtime.h>
#include <cstdint>

#define B_ 128
#define T_ 1024
#define C_ 256
#define NTHREADS 1024
#define NEG_INF (-3.402823466e38f)

typedef unsigned int v4u __attribute__((ext_vector_type(4)));
typedef int          v8i __attribute__((ext_vector_type(8)));
typedef int          v4i __attribute__((ext_vector_type(4)));

// LDS layout (floats): trans[65536] | state[256] | redm[1024] | redi[1024] | tags[1024]
#define LDS_FLOATS (65536 + 256 + 1024 + 1024 + 1024)

extern "C" __global__ __launch_bounds__(NTHREADS, 1)
void crf_viterbi_kernel(const float* __restrict__ inp,   // [B,T,C]
                        const float* __restrict__ trg,   // [C,C]
                        const int*   __restrict__ slen,  // [B,1]
                        float*       __restrict__ out,   // [B,T,C] one-hot
                        unsigned char* __restrict__ bp)  // [B,T,C] backpointers (ws)
{
    extern __shared__ float lds[];
    float* trans = lds;                                   // 256 KB
    float* state = lds + 65536;                           // 1 KB
    float* redm  = lds + 65536 + 256;                     // 4 KB
    int*   redi  = (int*)(lds + 65536 + 256 + 1024);      // 4 KB
    int*   tags  = (int*)(lds + 65536 + 256 + 2048);      // 4 KB

    const int tid = threadIdx.x;
    const int b   = blockIdx.x;
    const int q   = tid >> 8;     // 0..3 : which 64-wide slice of i
    const int j   = tid & 255;    // output tag column
    const int len = slen[b];

    // ---- Stage the 256x256 fp32 transition matrix into LDS via the TDM ----
#if __has_builtin(__builtin_amdgcn_tensor_load_to_lds) && __has_builtin(__builtin_amdgcn_s_wait_tensorcnt)
    if (tid < 32) {   // one wave issues the DMA descriptor
        uint64_t ga     = (uint64_t)(uintptr_t)trg;
        uint32_t ldsOff = (uint32_t)(uintptr_t)trans; // LDS aperture: low 32 bits == LDS byte offset
        v4u g0;
        g0.x = 1u;                                     // count = 1 valid descriptor
        g0.y = ldsOff;                                 // lds_addr
        g0.z = (uint32_t)ga;                           // global_addr[31:0]
        g0.w = (uint32_t)(ga >> 32) | (2u << 30);      // global_addr[56:32] | type=2 (image)
        v8i g1;
        g1[0] = (2 << 16);           // workgroup_mask=0, data_size=2 (4 bytes)
        g1[1] = (int)(256u << 16);   // tensor_dim0 = 256 (bits 79:48, low half)
        g1[2] = (int)(256u << 16);   // tensor_dim0 hi = 0 ; tensor_dim1 = 256 (low half)
        g1[3] = (int)(256u << 16);   // tensor_dim1 hi = 0 ; tile_dim0 = 256
        g1[4] = 256;                 // tile_dim1 = 256 ; tile_dim2 = 0
        g1[5] = 256;                 // tensor_dim0_stride = 256 (low 32)
        g1[6] = 0;                   // stride hi ; tensor_dim1_stride low16
        g1[7] = 1;                   // tensor_dim1_stride = 65536 (unused for 2D tile)
        v4i z4 = {0, 0, 0, 0};       // groups 2/3 unused (<=2D tensor)
        v8i z8 = {0, 0, 0, 0, 0, 0, 0, 0};
        __builtin_amdgcn_tensor_load_to_lds(g0, g1, z4, z4, z8, 0);  // 6-arg form (clang-23)
        __builtin_amdgcn_s_wait_tensorcnt(0);
    }
#else
    for (int idx = tid; idx < (C_ * C_) / 4; idx += NTHREADS)
        ((float4*)trans)[idx] = ((const float4*)trg)[idx];
#endif

    if (tid < C_) state[tid] = inp[((size_t)b * T_) * C_ + tid];   // init_state = inputs[:,0,:]
    __syncthreads();

    // ---------------- Forward Viterbi scan (sequential over t) ----------------
    for (int t = 1; t < T_; ++t) {
        if (t < len) {
            // issue next-step prefetch + this step's potentials early to hide latency
            __builtin_prefetch(inp + ((size_t)b * T_ + t + 1) * C_ + j, 0, 0);
            float pot = inp[((size_t)b * T_ + t) * C_ + j];

            const float* col = trans + (size_t)(q * 64) * C_ + j;
            const float* st  = state + q * 64;
            float m = NEG_INF; int mi = 0;
            #pragma unroll
            for (int k = 0; k < 64; ++k) {
                float s = st[k] + col[(size_t)k * C_];   // bank-conflict-free across lanes
                if (s > m) { m = s; mi = q * 64 + k; }   // strict > keeps smallest i (argmax ties)
            }
            redm[tid] = m;
            redi[tid] = mi;
            __syncthreads();
            if (tid < C_) {
                float bm = redm[j]; int bi = redi[j];
                #pragma unroll
                for (int qq = 1; qq < 4; ++qq) {
                    float vm = redm[qq * 256 + j];
                    int   vi = redi[qq * 256 + j];
                    if (vm > bm) { bm = vm; bi = vi; }   // ties keep lower q -> smaller i
                }
                state[j] = bm + pot;
                bp[((size_t)b * T_ + t) * C_ + j] = (unsigned char)bi;
            }
            __syncthreads();
        } else {
            // t >= seq_len: state frozen, backpointer = identity
            if (tid < C_)
                bp[((size_t)b * T_ + t) * C_ + j] = (unsigned char)j;
        }
    }

    // make this block's bp stores visible before the in-block pointer chase
    __threadfence();
    __syncthreads();

    // ---------------- Backtrack (per-batch sequential chase, L2-resident) ----------------
    if (tid == 0) {
        float bm = state[0]; int bi = 0;
        for (int c = 1; c < C_; ++c) {
            float v = state[c];
            if (v > bm) { bm = v; bi = c; }
        }
        int cur = bi;
        tags[T_ - 1] = cur;
        for (int t = T_ - 1; t >= 1; --t) {
            cur = bp[((size_t)b * T_ + t) * C_ + cur];
            tags[t - 1] = cur;
        }
    }
    __syncthreads();

    // ---------------- One-hot expansion (coalesced float4 stream) ----------------
    const int row = tid >> 6;          // 16 rows per iteration
    const int c4  = (tid & 63) * 4;
    for (int it = 0; it < T_ / 16; ++it) {
        int t  = it * 16 + row;
        int tg = tags[t];
        float4 v;
        v.x = (c4 + 0 == tg) ? 1.0f : 0.0f;
        v.y = (c4 + 1 == tg) ? 1.0f : 0.0f;
        v.z = (c4 + 2 == tg) ? 1.0f : 0.0f;
        v.w = (c4 + 3 == tg) ? 1.0f : 0.0f;
        *reinterpret_cast<float4*>(out + ((size_t)b * T_ + t) * C_ + c4) = v;
    }
}

extern "C" void kernel_launch(void* const* d_in, const int* in_sizes, int n_in,
                              void* d_out, int out_size, void* d_ws, size_t ws_size,
                              hipStream_t stream) {
    const float* inp  = (const float*)d_in[0];   // [128,1024,256] f32
    const float* trg  = (const float*)d_in[1];   // [256,256] f32
    const int*   slen = (const int*)d_in[2];     // [128,1] i32
    float*         out = (float*)d_out;
    unsigned char* bp  = (unsigned char*)d_ws;   // needs B*T*C = 32 MiB

    const size_t smem = (size_t)LDS_FLOATS * sizeof(float);   // ~269 KB (< 320 KB WGP LDS)
    (void)hipFuncSetAttribute(reinterpret_cast<const void*>(crf_viterbi_kernel),
                              hipFuncAttributeMaxDynamicSharedMemorySize, (int)smem);
    crf_viterbi_kernel<<<B_, NTHREADS, smem, stream>>>(inp, trg, slen, out, bp);
}